// HierarchicalSoftmax_50225347559810
// MI455X (gfx1250) — compile-verified
//
#include <hip/hip_runtime.h>
#include <math.h>

// ---------------- problem constants (from reference) ----------------
#define BT      2048          // B*S tokens
#define DMODEL  2048
#define DITEM   384
#define DCLRAW  118
#define NITEMS  100000
#define NCLUST  1024
#define MCLUST  128
#define KPAD    128           // padded cluster-raw dim (118 data + 1 bias + 9 zero)

typedef __attribute__((ext_vector_type(2))) float v2f;
typedef __attribute__((ext_vector_type(8))) float v8f;

// --------------------------------------------------------------------
// fp32 WMMA GEMM:  C[M x N] = A[M x K] * B[N x K]^T   (both K-major, NT)
// One wave (32 lanes) computes a 32x32 tile as 2x2 grid of 16x16 WMMA
// tiles -> 4 loads : 4 WMMAs per K-step (vs 8:4 for a 16x16 tile).
// A operand layout: lane L holds row M=L%16, VGPR0/1 = K = 2*(L/16)+0/1
// B operand layout: lane L holds col N=L%16, same K mapping
// C/D layout: vgpr r -> M = r + 8*(L/16), N = L%16
// --------------------------------------------------------------------
__device__ __forceinline__ v8f wmma4(v2f a, v2f b, v8f c) {
    return __builtin_amdgcn_wmma_f32_16x16x4_f32(
        /*neg_a=*/false, a, /*neg_b=*/false, b,
        /*c_mod=*/(short)0, c, /*reuse_a=*/false, /*reuse_b=*/false);
}

__device__ __forceinline__ void store_tile(float* __restrict__ C, int ldc,
                                           const v8f& acc, int rbase, int col,
                                           int N, int lane) {
    if (col < N) {
        float* cp = C + (size_t)(rbase + ((lane >> 4) << 3)) * ldc + col;
#pragma unroll
        for (int r = 0; r < 8; ++r) cp[(size_t)r * ldc] = acc[r];
    }
}

__global__ __launch_bounds__(32)
void gemm_nt_f32_wmma(const float* __restrict__ A, const float* __restrict__ B,
                      float* __restrict__ C, int M, int N, int K, int ldc) {
    const int lane = threadIdx.x;
    const int l16  = lane & 15;
    const int koff = (lane >> 4) << 1;          // 0 or 2
    const int tm   = blockIdx.x;                // 32-row tile index
    const int tn   = blockIdx.y;                // 32-col tile index

    const int r0 = tm * 32 + l16;               // M always a multiple of 32 here
    const int r1 = r0 + 16;
    const int c0 = tn * 32 + l16;
    const int c1 = c0 + 16;
    // clamp reads: column n of C depends only on row n of B, so OOB columns
    // may read any valid row; their results are simply never stored.
    const int bc0 = (c0 < N) ? c0 : (N - 1);
    const int bc1 = (c1 < N) ? c1 : (N - 1);

    const float* __restrict__ a0 = A + (size_t)r0  * K + koff;
    const float* __restrict__ a1 = A + (size_t)r1  * K + koff;
    const float* __restrict__ b0 = B + (size_t)bc0 * K + koff;
    const float* __restrict__ b1 = B + (size_t)bc1 * K + koff;

    v8f acc00 = {}, acc01 = {}, acc10 = {}, acc11 = {};
#pragma unroll 2
    for (int k = 0; k < K; k += 4) {
        v2f av0 = *(const v2f*)(a0 + k);
        v2f av1 = *(const v2f*)(a1 + k);
        v2f bv0 = *(const v2f*)(b0 + k);
        v2f bv1 = *(const v2f*)(b1 + k);
        acc00 = wmma4(av0, bv0, acc00);
        acc01 = wmma4(av0, bv1, acc01);
        acc10 = wmma4(av1, bv0, acc10);
        acc11 = wmma4(av1, bv1, acc11);
    }

    store_tile(C, ldc, acc00, tm * 32,      c0, N, lane);
    store_tile(C, ldc, acc01, tm * 32,      c1, N, lane);
    store_tile(C, ldc, acc10, tm * 32 + 16, c0, N, lane);
    store_tile(C, ldc, acc11, tm * 32 + 16, c1, N, lane);
}

// --------------------------------------------------------------------
// pad cluster_embeddings_raw [1024,118] -> CRpad [1024,128]
// col 118 = 1.0 (bias column), cols 119..127 = 0
// --------------------------------------------------------------------
__global__ void pad_craw_kernel(const float* __restrict__ raw, float* __restrict__ crp) {
    int c = blockIdx.x, j = threadIdx.x;        // block 128
    float v = 0.f;
    if (j < DCLRAW)       v = raw[(size_t)c * DCLRAW + j];
    else if (j == DCLRAW) v = 1.0f;
    crp[(size_t)c * KPAD + j] = v;
}

// --------------------------------------------------------------------
// per-token bias dots:  HPC[t,118] = h.b_cluster  (pairs with CRpad col 118),
// HPC[t,119..127] = 0, hbi[t] = h.b_item
// --------------------------------------------------------------------
__global__ void bias_dots_kernel(const float* __restrict__ H,
                                 const float* __restrict__ bclu,
                                 const float* __restrict__ bitm,
                                 float* __restrict__ HPC,
                                 float* __restrict__ hbi) {
    const int t = blockIdx.x, tid = threadIdx.x;   // block 256
    const float* h = H + (size_t)t * DMODEL;
    float sc = 0.f, si = 0.f;
    for (int d = tid; d < DMODEL; d += 256) {
        float hv = h[d];
        sc += hv * bclu[d];
        si += hv * bitm[d];
    }
    __shared__ float r1[256], r2[256];
    r1[tid] = sc; r2[tid] = si; __syncthreads();
    for (int o = 128; o > 0; o >>= 1) {
        if (tid < o) { r1[tid] += r1[tid + o]; r2[tid] += r2[tid + o]; }
        __syncthreads();
    }
    if (tid == 0) { HPC[(size_t)t * KPAD + DCLRAW] = r1[0]; hbi[t] = r2[0]; }
    if (tid > DCLRAW && tid < KPAD) HPC[(size_t)t * KPAD + tid] = 0.f;
}

// --------------------------------------------------------------------
// cluster-level log-softmax over 1024 logits per token:
//   tclp[t] = CL[t, cid] - lse,  accf[t] = (argmax == cid)
// --------------------------------------------------------------------
__global__ void cluster_softmax_kernel(const float* __restrict__ CL,
                                       const int* __restrict__ targets,
                                       const int* __restrict__ assign,
                                       float* __restrict__ tclp,
                                       float* __restrict__ accf) {
    const int t = blockIdx.x, tid = threadIdx.x;   // block 256
    const float* row = CL + (size_t)t * NCLUST;

    float m = -INFINITY; int mi = 0;
    for (int c = tid; c < NCLUST; c += 256) {
        float v = row[c];
        if (v > m) { m = v; mi = c; }
    }
    __shared__ float sv[256]; __shared__ int sidx[256];
    sv[tid] = m; sidx[tid] = mi; __syncthreads();
    for (int o = 128; o > 0; o >>= 1) {
        if (tid < o) {
            if (sv[tid + o] > sv[tid] ||
                (sv[tid + o] == sv[tid] && sidx[tid + o] < sidx[tid])) {
                sv[tid] = sv[tid + o]; sidx[tid] = sidx[tid + o];
            }
        }
        __syncthreads();
    }
    const float gmax = sv[0];
    const int   garg = sidx[0];
    __syncthreads();

    float s = 0.f;
    for (int c = tid; c < NCLUST; c += 256) s += __expf(row[c] - gmax);
    sv[tid] = s; __syncthreads();
    for (int o = 128; o > 0; o >>= 1) {
        if (tid < o) sv[tid] += sv[tid + o];
        __syncthreads();
    }
    if (tid == 0) {
        float lse = gmax + __logf(sv[0]);
        int cid = assign[targets[t]];
        tclp[t] = row[cid] - lse;
        accf[t] = (garg == cid) ? 1.f : 0.f;
    }
}

// --------------------------------------------------------------------
// item level: per token, gather <=128 member rows of item_embeddings,
// logits[m] = HPI[t,:].E[idx,:] + hbi[t]; masked log-softmax; pick target.
// block = 128 threads (one member per thread). The HPI row is staged into
// LDS via CDNA5 async global->LDS copies (ASYNCcnt / s_wait_asynccnt).
// --------------------------------------------------------------------
__global__ void item_level_kernel(const float* __restrict__ HPI,
                                  const float* __restrict__ E,
                                  const float* __restrict__ hbi,
                                  const int* __restrict__ targets,
                                  const int* __restrict__ assign,
                                  const int* __restrict__ cidx,
                                  float* __restrict__ tilp) {
    const int t = blockIdx.x, m = threadIdx.x;   // block 128
    __shared__ float hp[DITEM];
    __shared__ float red[MCLUST];
    __shared__ float s_tlp;

    const int tgt = targets[t];
    const int cid = assign[tgt];

    // async-stage HPI[t, :] into LDS (3 x b32 per lane, tracked by ASYNCcnt)
    {
        const float* src = HPI + (size_t)t * DITEM;
#pragma unroll
        for (int j = m; j < DITEM; j += MCLUST) {
            unsigned lds_off = (unsigned)(size_t)&hp[j];   // LDS byte offset
            const float* g = src + j;
            asm volatile("global_load_async_to_lds_b32 %0, %1, off"
                         :: "v"(lds_off), "v"(g) : "memory");
        }
        asm volatile("s_wait_asynccnt 0" ::: "memory");
    }
    if (m == 0) s_tlp = 0.f;
    __syncthreads();

    const int  idx   = cidx[(size_t)cid * MCLUST + m];
    const bool valid = (idx >= 0);
    const float* er  = E + (size_t)(valid ? idx : 0) * DITEM;

    float s = hbi[t];
#pragma unroll 8
    for (int j = 0; j < DITEM; j += 4) {
        float4 e = *(const float4*)(er + j);
        s += hp[j] * e.x + hp[j + 1] * e.y + hp[j + 2] * e.z + hp[j + 3] * e.w;
    }
    const float logit = valid ? s : -1e9f;

    red[m] = logit; __syncthreads();
    for (int o = 64; o > 0; o >>= 1) {
        if (m < o) red[m] = fmaxf(red[m], red[m + o]);
        __syncthreads();
    }
    const float gmax = red[0]; __syncthreads();
    red[m] = __expf(logit - gmax); __syncthreads();
    for (int o = 64; o > 0; o >>= 1) {
        if (m < o) red[m] += red[m + o];
        __syncthreads();
    }
    const float lse = gmax + __logf(red[0]);

    if (valid && idx == tgt) s_tlp = logit - lse;  // unique writer (members unique)
    __syncthreads();
    if (m == 0) tilp[t] = s_tlp;
}

// --------------------------------------------------------------------
// final masked reductions -> (total_loss, cluster_loss, item_loss, acc)
// --------------------------------------------------------------------
__global__ void finalize_kernel(const float* __restrict__ mask,
                                const float* __restrict__ tclp,
                                const float* __restrict__ tilp,
                                const float* __restrict__ accf,
                                float* __restrict__ out) {
    const int tid = threadIdx.x;   // single block of 256
    float sm = 0.f, sc = 0.f, si = 0.f, sa = 0.f;
    for (int t = tid; t < BT; t += 256) {
        float w = mask[t];
        sm += w;
        sc += tclp[t] * w;
        si += tilp[t] * w;
        sa += accf[t] * w;
    }
    __shared__ float a[256], b[256], c[256], d[256];
    a[tid] = sm; b[tid] = sc; c[tid] = si; d[tid] = sa; __syncthreads();
    for (int o = 128; o > 0; o >>= 1) {
        if (tid < o) {
            a[tid] += a[tid + o]; b[tid] += b[tid + o];
            c[tid] += c[tid + o]; d[tid] += d[tid + o];
        }
        __syncthreads();
    }
    if (tid == 0) {
        float denom = a[0] + 1e-8f;
        out[0] = -(b[0] + c[0]) / denom;
        out[1] = -b[0] / denom;
        out[2] = -c[0] / denom;
        out[3] =  d[0] / denom;
    }
}

// --------------------------------------------------------------------
extern "C" void kernel_launch(void* const* d_in, const int* in_sizes, int n_in,
                              void* d_out, int out_size, void* d_ws, size_t ws_size,
                              hipStream_t stream) {
    const float* H     = (const float*)d_in[0];   // [BT, 2048]
    const float* E     = (const float*)d_in[1];   // [100000, 384]
    const float* mask  = (const float*)d_in[2];   // [BT]
    const float* Wc    = (const float*)d_in[3];   // [118, 2048]
    const float* bc    = (const float*)d_in[4];   // [2048]
    const float* Wi    = (const float*)d_in[5];   // [384, 2048]
    const float* bi    = (const float*)d_in[6];   // [2048]
    const float* craw  = (const float*)d_in[7];   // [1024, 118]
    const int*   tgts  = (const int*)d_in[8];     // [BT]
    const int*   asg   = (const int*)d_in[9];     // [100000]
    const int*   cidx  = (const int*)d_in[10];    // [1024, 128]
    float* out = (float*)d_out;

    // workspace layout (fp32)
    float* HPC  = (float*)d_ws;                   // [BT, 128]  H@Wc^T + bias col
    float* CRP  = HPC + (size_t)BT * KPAD;        // [1024, 128] padded raw
    float* CL   = CRP + (size_t)NCLUST * KPAD;    // [BT, 1024] cluster logits
    float* HPI  = CL  + (size_t)BT * NCLUST;      // [BT, 384]  H@Wi^T
    float* hbi  = HPI + (size_t)BT * DITEM;       // [BT]
    float* tclp = hbi + BT;                       // [BT]
    float* accf = tclp + BT;                      // [BT]
    float* tilp = accf + BT;                      // [BT]

    // 1) pad cluster raw embeddings (adds bias column of ones)
    pad_craw_kernel<<<NCLUST, 128, 0, stream>>>(craw, CRP);

    // 2) per-token bias dots (fills HPC cols 118..127, hbi)
    bias_dots_kernel<<<BT, 256, 0, stream>>>(H, bc, bi, HPC, hbi);

    // 3) HPC[:, 0:118] = H @ Wc^T        (M=2048, N=118, K=2048)
    gemm_nt_f32_wmma<<<dim3(BT / 32, (DCLRAW + 31) / 32), 32, 0, stream>>>(
        H, Wc, HPC, BT, DCLRAW, DMODEL, KPAD);

    // 4) HPI = H @ Wi^T                  (M=2048, N=384, K=2048)
    gemm_nt_f32_wmma<<<dim3(BT / 32, DITEM / 32), 32, 0, stream>>>(
        H, Wi, HPI, BT, DITEM, DMODEL, DITEM);

    // 5) CL = HPC @ CRP^T                (M=2048, N=1024, K=128) — bias included
    gemm_nt_f32_wmma<<<dim3(BT / 32, NCLUST / 32), 32, 0, stream>>>(
        HPC, CRP, CL, BT, NCLUST, KPAD, NCLUST);

    // 6) cluster-level log-softmax + accuracy flag
    cluster_softmax_kernel<<<BT, 256, 0, stream>>>(CL, tgts, asg, tclp, accf);

    // 7) item-level gathered logits + masked log-softmax
    item_level_kernel<<<BT, MCLUST, 0, stream>>>(HPI, E, hbi, tgts, asg, cidx, tilp);

    // 8) masked reductions -> 4 scalars
    finalize_kernel<<<1, 256, 0, stream>>>(mask, tclp, tilp, accf, out);
}